// CAM_58798102282729
// MI455X (gfx1250) — compile-verified
//
#include <hip/hip_runtime.h>
#include <stdint.h>

typedef float v2f __attribute__((ext_vector_type(2)));
typedef float v8f __attribute__((ext_vector_type(8)));

#define CS   262144   // 64^3 channel stride (full res)
#define DSN  32768    // 32^3 channel stride (downsampled)

// ---------------------------------------------------------------------------
// WMMA helpers: apply a 4x4 channel-mix matrix to 16-voxel tiles.
// D(16x16) = A(16x4) x B(4x16) + C(16x16)  via V_WMMA_F32_16X16X4_F32.
// A layout (ISA 7.12.2, 32-bit A 16x4): lanes 0-15 rows M=0..15 with K={0,1}
// in VGPR0/1; lanes 16-31 same rows with K={2,3}.
// B layout (4x16, mirrored): VGPR0 = K0 (lanes<16) / K2 (lanes>=16),
// VGPR1 = K1 / K3, N = lane&15.
// C/D: VGPR r = row M=r (lanes 0-15) / M=r+8 (lanes 16-31), N = lane&15.
// ---------------------------------------------------------------------------
__device__ inline v2f afrag_mat(const float* __restrict__ M) {
  int lane = threadIdx.x & 31;
  int row  = lane & 15;
  int k    = (lane & 16) ? 2 : 0;
  v2f a;
  a.x = (row < 4) ? M[row * 4 + k]     : 0.0f;
  a.y = (row < 4) ? M[row * 4 + k + 1] : 0.0f;
  return a;
}

__device__ inline v2f afrag_mat_sw(const float* __restrict__ M,
                                   const float* __restrict__ sw) {
  int lane = threadIdx.x & 31;
  int row  = lane & 15;
  int k    = (lane & 16) ? 2 : 0;
  v2f a;
  a.x = (row < 4) ? M[row * 4 + k]     * sw[k]     : 0.0f;
  a.y = (row < 4) ? M[row * 4 + k + 1] * sw[k + 1] : 0.0f;
  return a;
}

// Each lane owns channel values s0..s3 for voxel (tileBase + lane); the wave's
// 32 voxels form two 16-wide N tiles: [tb0..tb0+15] (lanes 0-15) and
// [tb1..tb1+15] (lanes 16-31). out[ch][v] = sum_k A[ch][k]*s_k[v] + cb[ch][v].
__device__ inline void mix_store2(v2f a, float s0, float s1, float s2, float s3,
                                  const float* __restrict__ cb,
                                  float* __restrict__ out, int tb0, int tb1) {
  int lane = threadIdx.x & 31;
  int col  = lane & 15;
  bool lo  = lane < 16;
#pragma unroll
  for (int t = 0; t < 2; ++t) {
    int src  = col + t * 16;                    // lane holding voxel col of tile t
    float u0 = __shfl(s0, src, 32);
    float u1 = __shfl(s1, src, 32);
    float u2 = __shfl(s2, src, 32);
    float u3 = __shfl(s3, src, 32);
    v2f b;
    b.x = lo ? u0 : u2;                          // K0 / K2
    b.y = lo ? u1 : u3;                          // K1 / K3
    int v = (t ? tb1 : tb0) + col;
    v8f c = {0.f, 0.f, 0.f, 0.f, 0.f, 0.f, 0.f, 0.f};
    if (lo) {                                    // rows 0..3 valid, rest zero
      c.s0 = cb[v]; c.s1 = cb[CS + v]; c.s2 = cb[2 * CS + v]; c.s3 = cb[3 * CS + v];
    }
    v8f d = __builtin_amdgcn_wmma_f32_16x16x4_f32(false, a, false, b,
                                                  (short)0, c, false, false);
    if (lo) {
      out[v] = d.s0; out[CS + v] = d.s1; out[2 * CS + v] = d.s2; out[3 * CS + v] = d.s3;
    }
  }
}

// ---------------------------------------------------------------------------
// Stage 0: pack imgn = img * smooth_norm as voxel-interleaved float2 so the
// smooth-message halo can be fetched with async B64 copies straight to LDS.
// ---------------------------------------------------------------------------
__global__ void k_imgpack(const float* __restrict__ img,
                          const float* __restrict__ snorm,
                          float2* __restrict__ im2) {
  int v = blockIdx.x * 256 + threadIdx.x;
  float sn = snorm[0];
  im2[v] = make_float2(img[v] * sn, img[CS + v] * sn);
}

// ---------------------------------------------------------------------------
// Stage 1: trilinear downsample 64^3 -> 32^3 of img*pn, atlas*pn, atlas_label
// (reference resize: pos = i * (n-1)/(out-1), clamp, lerp; separable == 8-tap)
// ---------------------------------------------------------------------------
__global__ void k_down(const float* __restrict__ img, const float* __restrict__ atlas,
                       const float* __restrict__ albl, const float* __restrict__ pnorm,
                       float* __restrict__ f1, float* __restrict__ f2,
                       float* __restrict__ alds) {
  int o = blockIdx.x * 256 + threadIdx.x;        // 32^3 voxels
  int ox = o >> 10, oy = (o >> 5) & 31, oz = o & 31;
  const float sc = 63.0f / 31.0f;
  float px = ox * sc, py = oy * sc, pz = oz * sc;
  int x0 = min((int)px, 63), y0 = min((int)py, 63), z0 = min((int)pz, 63);
  int x1 = min(x0 + 1, 63), y1 = min(y0 + 1, 63), z1 = min(z0 + 1, 63);
  float wx = px - x0, wy = py - y0, wz = pz - z0;
  float pn = pnorm[0];
  float a0 = 0, a1 = 0, b0 = 0, b1 = 0, l0 = 0, l1 = 0, l2 = 0, l3 = 0;
#pragma unroll
  for (int cx = 0; cx < 2; ++cx)
#pragma unroll
    for (int cy = 0; cy < 2; ++cy)
#pragma unroll
      for (int cz = 0; cz < 2; ++cz) {
        int X = cx ? x1 : x0, Y = cy ? y1 : y0, Z = cz ? z1 : z0;
        float W = (cx ? wx : 1.f - wx) * (cy ? wy : 1.f - wy) * (cz ? wz : 1.f - wz);
        int idx = ((X << 6) + Y) * 64 + Z;
        a0 = fmaf(W, img[idx], a0);        a1 = fmaf(W, img[CS + idx], a1);
        b0 = fmaf(W, atlas[idx], b0);      b1 = fmaf(W, atlas[CS + idx], b1);
        const float* lp = albl + (size_t)idx * 4;   // (x,y,z,c) layout
        l0 = fmaf(W, lp[0], l0); l1 = fmaf(W, lp[1], l1);
        l2 = fmaf(W, lp[2], l2); l3 = fmaf(W, lp[3], l3);
      }
  f1[o] = a0 * pn; f1[DSN + o] = a1 * pn;
  f2[o] = b0 * pn; f2[DSN + o] = b1 * pn;
  alds[o] = l0; alds[DSN + o] = l1; alds[2 * DSN + o] = l2; alds[3 * DSN + o] = l3;
}

// ---------------------------------------------------------------------------
// Stage 2: prior message at 32^3:
// pm[c,i] = sum_d al[c,i+d] * exp(-0.5 * sum_ch (f1[ch,i]-f2[ch,i+d])^2)
// ---------------------------------------------------------------------------
__global__ void k_priormsg(const float* __restrict__ f1, const float* __restrict__ f2,
                           const float* __restrict__ alds, float* __restrict__ pmds) {
  int i = blockIdx.x * 256 + threadIdx.x;
  int ix = i >> 10, iy = (i >> 5) & 31, iz = i & 31;
  float c0 = f1[i], c1 = f1[DSN + i];
  float m0 = 0, m1 = 0, m2 = 0, m3 = 0;
  for (int dx = -2; dx <= 2; ++dx) {
    int nx = ix + dx; if ((unsigned)nx >= 32u) continue;
    for (int dy = -2; dy <= 2; ++dy) {
      int ny = iy + dy; if ((unsigned)ny >= 32u) continue;
      for (int dz = -2; dz <= 2; ++dz) {
        int nz = iz + dz; if ((unsigned)nz >= 32u) continue;
        int j = ((nx << 5) + ny) * 32 + nz;
        float d0 = c0 - f2[j], d1 = c1 - f2[DSN + j];
        float w = __expf(-0.5f * (d0 * d0 + d1 * d1));
        m0 = fmaf(alds[j], w, m0);           m1 = fmaf(alds[DSN + j], w, m1);
        m2 = fmaf(alds[2 * DSN + j], w, m2); m3 = fmaf(alds[3 * DSN + j], w, m3);
      }
    }
  }
  pmds[i] = m0; pmds[DSN + i] = m1; pmds[2 * DSN + i] = m2; pmds[3 * DSN + i] = m3;
}

// ---------------------------------------------------------------------------
// Stage 3: upsample 32->64, apply prior_weight patch, then (WMMA)
// base = compat @ prior_term + unary
// ---------------------------------------------------------------------------
__global__ void __launch_bounds__(256)
k_prior_final(const float* __restrict__ pmds, const float* __restrict__ pwt,
              const int* __restrict__ ps, const float* __restrict__ unary,
              const float* __restrict__ compat, float* __restrict__ base) {
  int v = blockIdx.x * 256 + threadIdx.x;        // linear 64^3 voxel
  int x = v >> 12, y = (v >> 6) & 63, z = v & 63;
  const float sc = 31.0f / 63.0f;
  float px = x * sc, py = y * sc, pz = z * sc;
  int x0 = min((int)px, 31), y0 = min((int)py, 31), z0 = min((int)pz, 31);
  int x1 = min(x0 + 1, 31), y1 = min(y0 + 1, 31), z1 = min(z0 + 1, 31);
  float wx = px - x0, wy = py - y0, wz = pz - z0;
  float m0 = 0, m1 = 0, m2 = 0, m3 = 0;
#pragma unroll
  for (int cx = 0; cx < 2; ++cx)
#pragma unroll
    for (int cy = 0; cy < 2; ++cy)
#pragma unroll
      for (int cz = 0; cz < 2; ++cz) {
        int X = cx ? x1 : x0, Y = cy ? y1 : y0, Z = cz ? z1 : z0;
        float W = (cx ? wx : 1.f - wx) * (cy ? wy : 1.f - wy) * (cz ? wz : 1.f - wz);
        int j = ((X << 5) + Y) * 32 + Z;
        m0 = fmaf(W, pmds[j], m0);           m1 = fmaf(W, pmds[DSN + j], m1);
        m2 = fmaf(W, pmds[2 * DSN + j], m2); m3 = fmaf(W, pmds[3 * DSN + j], m3);
      }
  int sx = min(max(ps[0], 0), 32), sy = min(max(ps[1], 0), 32), sz = min(max(ps[2], 0), 32);
  float pw = pwt[((sx + x) * 96 + (sy + y)) * 96 + (sz + z)];
  m0 *= pw; m1 *= pw; m2 *= pw; m3 *= pw;

  v2f a = afrag_mat(compat);
  int wb = blockIdx.x * 256 + (threadIdx.x & ~31);   // wave's first voxel (linear)
  mix_store2(a, m0, m1, m2, m3, unary, base, wb, wb + 16);
}

// ---------------------------------------------------------------------------
// Stage 4a: softmax over 4 channels per voxel; output voxel-interleaved float4
// so the message kernel can async-copy contiguous rows to LDS.
// ---------------------------------------------------------------------------
__global__ void k_softmax(const float* __restrict__ q, float4* __restrict__ p4) {
  int v = blockIdx.x * 256 + threadIdx.x;
  float q0 = q[v], q1 = q[CS + v], q2 = q[2 * CS + v], q3 = q[3 * CS + v];
  float m = fmaxf(fmaxf(q0, q1), fmaxf(q2, q3));
  float e0 = __expf(q0 - m), e1 = __expf(q1 - m), e2 = __expf(q2 - m), e3 = __expf(q3 - m);
  float r = 1.0f / (e0 + e1 + e2 + e3);
  p4[v] = make_float4(e0 * r, e1 * r, e2 * r, e3 * r);
}

// ---------------------------------------------------------------------------
// Stage 4b: LDS-tiled 125-tap smooth message + WMMA channel mix:
// q = base + (compat * diag(sw)) @ sm
// Tile: 2x8x16 voxels per 256-thread block, halo 2 each side (6x12x20 cells).
// Halo fill uses GLOBAL_LOAD_ASYNC_TO_LDS_B128/B64 (ASYNCcnt) for in-bounds
// cells; OOB cells are zeroed with plain DS stores (p=0 kills contribution,
// img=0 keeps the Gaussian weight finite).
// ---------------------------------------------------------------------------
#define HXD 6
#define HYD 12
#define HZD 20
#define HN  (HXD * HYD * HZD)   // 1440 cells

__global__ void __launch_bounds__(256)
k_smooth(const float4* __restrict__ p4, const float2* __restrict__ im2,
         const float* __restrict__ sw, const float* __restrict__ compat,
         const float* __restrict__ base, float* __restrict__ out) {
  __shared__ float4 sp[HN];   // 4 prob channels      (23.0 KB)
  __shared__ float2 si[HN];   // 2 img*norm channels  (11.5 KB)
  int tid = threadIdx.x;
  int tz = blockIdx.x, ty = blockIdx.y, tx = blockIdx.z;

  for (int h = tid; h < HN; h += 256) {
    int hx = h / (HYD * HZD);
    int r  = h - hx * (HYD * HZD);
    int hy = r / HZD;
    int hz = r - hy * HZD;
    int gx = tx * 2 + hx - 2, gy = ty * 8 + hy - 2, gz = tz * 16 + hz - 2;
    if ((unsigned)gx < 64u && (unsigned)gy < 64u && (unsigned)gz < 64u) {
      int idx = ((gx << 6) + gy) * 64 + gz;
      // LDS byte offsets: addrspace(3) offsets == low 32 bits of generic addr
      uint32_t lds_p = (uint32_t)(uintptr_t)(&sp[h]);
      uint32_t lds_i = (uint32_t)(uintptr_t)(&si[h]);
      uint64_t ga_p  = (uint64_t)(uintptr_t)(p4 + idx);
      uint64_t ga_i  = (uint64_t)(uintptr_t)(im2 + idx);
      asm volatile("global_load_async_to_lds_b128 %0, %1, off"
                   :: "v"(lds_p), "v"(ga_p) : "memory");
      asm volatile("global_load_async_to_lds_b64 %0, %1, off"
                   :: "v"(lds_i), "v"(ga_i) : "memory");
    } else {
      sp[h] = make_float4(0.f, 0.f, 0.f, 0.f);
      si[h] = make_float2(0.f, 0.f);
    }
  }

  // Prefetch the base rows this thread's wave will need after the 125-tap loop.
  int lz = tid & 15, ly = (tid >> 4) & 7, lx = tid >> 7;
  int vg = (((tx * 2 + lx) * 64) + (ty * 8 + ly)) * 64 + (tz * 16 + lz);
  __builtin_prefetch(base + vg);
  __builtin_prefetch(base + CS + vg);
  __builtin_prefetch(base + 2 * CS + vg);
  __builtin_prefetch(base + 3 * CS + vg);

  asm volatile("s_wait_asynccnt 0x0" ::: "memory");   // async LDS writes visible
  __syncthreads();                                    // (DS zero-fills via barrier)

  int hc = ((lx + 2) * HYD + (ly + 2)) * HZD + (lz + 2);
  float2 cc = si[hc];
  float m0 = 0, m1 = 0, m2 = 0, m3 = 0;
  for (int dx = -2; dx <= 2; ++dx) {
#pragma unroll
    for (int dy = -2; dy <= 2; ++dy) {
      int hb = hc + dx * (HYD * HZD) + dy * HZD;
#pragma unroll
      for (int dz = -2; dz <= 2; ++dz) {
        int h = hb + dz;
        float2 iv = si[h];
        float d0 = iv.x - cc.x, d1 = iv.y - cc.y;
        float w = __expf(-0.5f * (d0 * d0 + d1 * d1));
        float4 pv = sp[h];
        m0 = fmaf(pv.x, w, m0); m1 = fmaf(pv.y, w, m1);
        m2 = fmaf(pv.z, w, m2); m3 = fmaf(pv.w, w, m3);
      }
    }
  }

  // wave-uniform tile bases: lanes 0-15 -> row (wly), lanes 16-31 -> row (wly+1)
  v2f a = afrag_mat_sw(compat, sw);
  int wt  = tid & ~31;
  int wly = (wt >> 4) & 7;
  int wlx = wt >> 7;
  int tb0 = (((tx * 2 + wlx) * 64) + (ty * 8 + wly)) * 64 + tz * 16;
  mix_store2(a, m0, m1, m2, m3, base, out, tb0, tb0 + 64);
}

// ---------------------------------------------------------------------------
// Launch
// ---------------------------------------------------------------------------
extern "C" void kernel_launch(void* const* d_in, const int* in_sizes, int n_in,
                              void* d_out, int out_size, void* d_ws, size_t ws_size,
                              hipStream_t stream) {
  (void)in_sizes; (void)n_in; (void)out_size; (void)ws_size;
  const float* unary  = (const float*)d_in[0];   // (1,4,64,64,64)
  const float* img    = (const float*)d_in[1];   // (1,2,64,64,64)
  const float* atlas  = (const float*)d_in[2];   // (1,2,64,64,64)
  const float* albl   = (const float*)d_in[3];   // (1,64,64,64,4)
  const int*   ps     = (const int*)d_in[4];     // (1,3)
  const float* snorm  = (const float*)d_in[5];   // scalar
  const float* sw     = (const float*)d_in[6];   // (4,)
  const float* pnorm  = (const float*)d_in[7];   // scalar
  const float* pwt    = (const float*)d_in[8];   // (96,96,96)
  const float* compat = (const float*)d_in[9];   // (4,4)
  float* out = (float*)d_out;                    // q, (1,4,64,64,64)

  float* ws    = (float*)d_ws;                   // ~12 MB total
  float*  f1   = ws;                             // 2 * 32^3
  float*  f2   = ws + 65536;                     // 2 * 32^3
  float*  alds = ws + 131072;                    // 4 * 32^3
  float*  pmds = ws + 262144;                    // 4 * 32^3
  float*  base = ws + 393216;                    // 4 * 64^3
  float4* p4   = (float4*)(ws + 1441792);        // 64^3 float4 (16B aligned)
  float2* im2  = (float2*)(ws + 2490368);        // 64^3 float2 (8B aligned)

  k_imgpack<<<1024, 256, 0, stream>>>(img, snorm, im2);
  k_down<<<128, 256, 0, stream>>>(img, atlas, albl, pnorm, f1, f2, alds);
  k_priormsg<<<128, 256, 0, stream>>>(f1, f2, alds, pmds);
  k_prior_final<<<1024, 256, 0, stream>>>(pmds, pwt, ps, unary, compat, base);

  for (int it = 0; it < 4; ++it) {
    k_softmax<<<1024, 256, 0, stream>>>(it == 0 ? unary : out, p4);
    k_smooth<<<dim3(4, 8, 32), 256, 0, stream>>>(p4, im2, sw, compat, base, out);
  }
}